// NablaT_52243982189155
// MI455X (gfx1250) — compile-verified
//
#include <hip/hip_runtime.h>
#include <hip/hip_bf16.h>
#include <stdint.h>

// NablaT (divergence adjoint), 3D, Dirichlet boundaries, on (3,320,320,320) f32.
// out[z,y,x] = (z>0? x0[z-1,y,x]:0) - (z<319? x0[z,y,x]:0)
//            + (y>0? x1[z,y-1,x]:0) - (y<319? x1[z,y,x]:0)
//            + (x>0? x2[z,y,x-1]:0) - (x<319? x2[z,y,x]:0)
//
// Bandwidth-bound: ~524 MB unique HBM traffic -> ~22.5us floor at 23.3 TB/s.
// CDNA5 paths: GLOBAL_LOAD_ASYNC_TO_LDS_B128 (ASYNCcnt, NT hint) stages x2 rows
// into LDS; s_wait_asynccnt split-counter wait; NT b128 store for the output.
// grid = (40, 320): blockIdx.y = z (z-predicates uniform), blockIdx.x = 8-row
// group. 320 threads/block, each thread owns two float4 quads (rows r, r+4).

typedef float v4f __attribute__((ext_vector_type(4)));

#define N  320
#define NN (N * N)
#define Q  80                 // float4 quads per row
#define ROWS 8                // rows per block; 320 % 8 == 0 -> never crosses a plane
#define TPB 320               // 10 wave32; each thread: 2 quads

__global__ __launch_bounds__(TPB)
void nablaT_kernel(const float* __restrict__ x, float* __restrict__ out)
{
    __shared__ float lds_x2[ROWS * N];   // 10 KB

    const float* x0 = x;
    const float* x1 = x + (size_t)NN * N;
    const float* x2 = x + (size_t)2 * NN * N;

    const int tid = threadIdx.x;
    const int r   = tid / Q;            // 0..3
    const int q   = tid - r * Q;        // 0..79

    const int z  = blockIdx.y;              // uniform per block
    const int y0 = blockIdx.x * ROWS + r;   // this thread's rows: y0 and y0+4
    // y1 = y0 + 4;  y0 <= 315 (so y0 < N-1 always), y1 >= 4 (so y1 > 0 always)

    const int o0  = (z * N + y0) * N + q * 4;
    const int o1  = o0 + 4 * N;
    const int li0 = r * N + q * 4;
    const int li1 = li0 + 4 * N;

    // ---- CDNA5 async global -> LDS staging of two x2 rows (ASYNCcnt, NT) ----
    unsigned int lo0 = (unsigned int)(uintptr_t)(&lds_x2[li0]);
    unsigned int lo1 = (unsigned int)(uintptr_t)(&lds_x2[li1]);
    unsigned long long ga0 = (unsigned long long)(uintptr_t)(x2 + o0);
    unsigned long long ga1 = (unsigned long long)(uintptr_t)(x2 + o1);
    asm volatile("global_load_async_to_lds_b128 %0, %1, off th:TH_LOAD_NT"
                 :: "v"(lo0), "v"(ga0) : "memory");
    asm volatile("global_load_async_to_lds_b128 %0, %1, off th:TH_LOAD_NT"
                 :: "v"(lo1), "v"(ga1) : "memory");

    // ---- overlap: unconditional b128 loads, boundaries via clamped address ----
    const bool za  = (z > 0);          // uniform
    const bool zb  = (z < N - 1);      // uniform
    const bool ya0 = (y0 > 0);         // false only for first row of plane
    const bool yb1 = (y0 + 4 < N - 1); // false only for last row of plane

    v4f a0 = *(const v4f*)(x0 + (za ? o0 - NN : o0));
    v4f b0 = *(const v4f*)(x0 + o0);
    v4f a1 = *(const v4f*)(x0 + (za ? o1 - NN : o1));
    v4f b1 = *(const v4f*)(x0 + o1);

    v4f c0 = *(const v4f*)(x1 + (ya0 ? o0 - N : o0));
    v4f d0 = *(const v4f*)(x1 + o0);
    v4f c1 = *(const v4f*)(x1 + o1 - N);   // y1-1 = y0+3 >= 3: always valid
    v4f d1 = *(const v4f*)(x1 + o1);

    const v4f z4 = {0.f, 0.f, 0.f, 0.f};
    a0 = za  ? a0 : z4;   a1 = za  ? a1 : z4;   // z==0:   drop prev-plane term
    b0 = zb  ? b0 : z4;   b1 = zb  ? b1 : z4;   // z==319: drop cur-plane term
    c0 = ya0 ? c0 : z4;                         // y0==0:  drop prev-row term
    d1 = yb1 ? d1 : z4;                         // y1==319: drop cur-row term
    // (d-row0 and c-row1 terms are always kept: y0 <= 315 < 319, 4 <= y1)

    // wait for this wave's async LDS writes, then sync all 10 waves
    asm volatile("s_wait_asynccnt 0x0" ::: "memory");
    __syncthreads();

    // ---- x-dimension backward difference from LDS (shifted access) ----
    v4f v0 = *(const v4f*)(&lds_x2[li0]);
    v4f v1 = *(const v4f*)(&lds_x2[li1]);
    const bool qa = (q > 0), qb = (q < Q - 1);
    float p0 = lds_x2[qa ? li0 - 1 : li0];
    float p1 = lds_x2[qa ? li1 - 1 : li1];
    p0 = qa ? p0 : 0.f;                 // x==0   -> prev term 0
    p1 = qa ? p1 : 0.f;
    float cw0 = qb ? v0.w : 0.f;        // x==319 -> cur term 0
    float cw1 = qb ? v1.w : 0.f;

    v4f r0, r1;
    r0.x = (a0.x - b0.x) + (c0.x - d0.x) + (p0   - v0.x);
    r0.y = (a0.y - b0.y) + (c0.y - d0.y) + (v0.x - v0.y);
    r0.z = (a0.z - b0.z) + (c0.z - d0.z) + (v0.y - v0.z);
    r0.w = (a0.w - b0.w) + (c0.w - d0.w) + (v0.z - cw0);

    r1.x = (a1.x - b1.x) + (c1.x - d1.x) + (p1   - v1.x);
    r1.y = (a1.y - b1.y) + (c1.y - d1.y) + (v1.x - v1.y);
    r1.z = (a1.z - b1.z) + (c1.z - d1.z) + (v1.y - v1.z);
    r1.w = (a1.w - b1.w) + (c1.w - d1.w) + (v1.z - cw1);

    __builtin_nontemporal_store(r0, (v4f*)(out + o0));
    __builtin_nontemporal_store(r1, (v4f*)(out + o1));
}

extern "C" void kernel_launch(void* const* d_in, const int* in_sizes, int n_in,
                              void* d_out, int out_size, void* d_ws, size_t ws_size,
                              hipStream_t stream) {
    (void)in_sizes; (void)n_in; (void)out_size; (void)d_ws; (void)ws_size;
    const float* xin = (const float*)d_in[0];   // (3,320,320,320) f32
    float* out = (float*)d_out;                 // (320,320,320) f32

    dim3 grid(N / ROWS, N, 1);                  // (40, 320)
    nablaT_kernel<<<grid, TPB, 0, stream>>>(xin, out);
}